// HighwayGRU_51333449121849
// MI455X (gfx1250) — compile-verified
//
#include <hip/hip_runtime.h>
#include <hip/hip_bf16.h>
#include <math.h>

typedef __attribute__((ext_vector_type(2))) float v2f;
typedef __attribute__((ext_vector_type(8))) float v8f;

#define S_LEN 2048
#define BATCH 64
#define IDIM  256
#define HDIM  256
#define WROW  512          // row stride of W matrices (I+H)
#define LSTR  260          // padded LDS row stride; 260 mod 64 == 4 -> conflict-free WMMA A loads

// --- force global (AS1) loads so hot loops emit global_load_*, not flat_load_* ---
typedef __attribute__((address_space(1))) const float gc_f32;
typedef __attribute__((address_space(1))) const v2f   gc_v2f;

__device__ __forceinline__ float gldf(const float* p) {
    return *(gc_f32*)(unsigned long long)p;
}
__device__ __forceinline__ v2f gldv2(const float* p) {
    return *(gc_v2f*)(unsigned long long)p;
}

__device__ __forceinline__ float sigmoid_f(float x) {
    return 1.0f / (1.0f + __expf(-x));   // exp overflow -> inf -> 0: correct limit, no NaN
}
__device__ __forceinline__ float tanh_f(float x) {
    float cx = fminf(fmaxf(x, -15.0f), 15.0f);
    float e  = __expf(2.0f * cx);
    return (e - 1.0f) / (e + 1.0f);
}

// ---------------------------------------------------------------------------
// Phase 1: proj[s][gate][b][h] = x[s,b,:] . W_gate_x[h,:] + b_gate[h]
// gate order: 0=z 1=r 2=c 3=g(Wh). One block per s (64 rows), 32 waves,
// 256 output tiles (4 m-tiles x 64 n-tiles across 4 gates), 8 tiles/wave.
// ---------------------------------------------------------------------------
__global__ __launch_bounds__(1024) void hgru_proj_kernel(
    const float* __restrict__ x,
    const float* __restrict__ Wz, const float* __restrict__ bz,
    const float* __restrict__ Wr, const float* __restrict__ br,
    const float* __restrict__ Wc, const float* __restrict__ bc,
    const float* __restrict__ Wh, const float* __restrict__ bh,
    float* __restrict__ proj)
{
    __shared__ float sx[BATCH * LSTR];
    const int tid  = threadIdx.x;
    const int s    = blockIdx.x;
    const int lane = tid & 31;
    const int wid  = tid >> 5;
    const int l16  = lane & 15;
    const int hi   = lane >> 4;

    // cooperative load x[s] (64x256 f32) into padded LDS
    {
        const float4* src = (const float4*)(x + (size_t)s * BATCH * IDIM);
        #pragma unroll
        for (int i = 0; i < 4; ++i) {
            int fl  = tid + i * 1024;          // float4 index 0..4095
            int row = (fl * 4) >> 8;
            int col = (fl * 4) & 255;
            *(float4*)&sx[row * LSTR + col] = src[fl];
        }
    }
    __syncthreads();

    #pragma unroll
    for (int i = 0; i < 8; ++i) {
        int t    = wid * 8 + i;                // 0..255
        int m0   = (t & 3) << 4;
        int nt   = t >> 2;                     // 0..63
        int gate = nt >> 4;
        int n0   = (nt & 15) << 4;
        const float* W = (gate == 0) ? Wz : (gate == 1) ? Wr : (gate == 2) ? Wc : Wh;
        const float* B = (gate == 0) ? bz : (gate == 1) ? br : (gate == 2) ? bc : bh;
        float bias = gldf(B + n0 + l16);
        v8f acc;
        #pragma unroll
        for (int v = 0; v < 8; ++v) acc[v] = bias;
        const float* arow = sx + (m0 + l16) * LSTR + 2 * hi;
        const float* brow = W  + (n0 + l16) * WROW + 2 * hi;    // x-part: cols [0,256)
        #pragma unroll 8
        for (int k0 = 0; k0 < IDIM; k0 += 4) {
            v2f a = *(const v2f*)(arow + k0);
            v2f b = gldv2(brow + k0);
            acc = __builtin_amdgcn_wmma_f32_16x16x4_f32(false, a, false, b,
                                                        (short)0, acc, false, false);
        }
        float* pb = proj + (((size_t)s * 4 + gate) * BATCH) * HDIM;
        #pragma unroll
        for (int v = 0; v < 8; ++v)
            pb[(m0 + v + 8 * hi) * HDIM + n0 + l16] = acc[v];
    }
}

// ---------------------------------------------------------------------------
// Phase 2: persistent single-workgroup recurrent scan. 1024 threads = 32 waves.
// Per step: matmul-1 (z,r,g: 192 tiles, 6/wave) -> barrier ->
//           matmul-2 (c: 64 tiles, 2/wave) + gate combine -> barrier ->
//           LayerNorm (2 rows/wave) + store -> barrier.
// ---------------------------------------------------------------------------
__global__ __launch_bounds__(1024) void hgru_scan_kernel(
    const float* __restrict__ proj,
    const float* __restrict__ Wz, const float* __restrict__ Wr,
    const float* __restrict__ Wc, const float* __restrict__ Wh,
    const float* __restrict__ gamma, const float* __restrict__ beta,
    float* __restrict__ out)
{
    __shared__ float sh_h [BATCH * LSTR];   // carried state h
    __shared__ float sh_rh[BATCH * LSTR];   // r*h (A operand of c-matmul)
    __shared__ float sh_z [BATCH * LSTR];   // z gate, later reused for h_final
    __shared__ float sh_g [BATCH * LSTR];   // g gate

    const int tid  = threadIdx.x;
    const int lane = tid & 31;
    const int wid  = tid >> 5;
    const int l16  = lane & 15;
    const int hi   = lane >> 4;

    for (int i = tid; i < BATCH * LSTR; i += 1024) sh_h[i] = 0.0f;

    float gam[8], bet[8];
    #pragma unroll
    for (int j = 0; j < 8; ++j) {
        gam[j] = gldf(gamma + lane * 8 + j);
        bet[j] = gldf(beta  + lane * 8 + j);
    }
    __syncthreads();

    for (int s = 0; s < S_LEN; ++s) {
        const float* projs = proj + (size_t)s * (4 * BATCH * HDIM);

        // prefetch next step's pre-activations (256KB) into cache
        if (s + 1 < S_LEN) {
            const float* pn = projs + 4 * BATCH * HDIM + tid * 64;
            __builtin_prefetch(pn,      0, 1);
            __builtin_prefetch(pn + 32, 0, 1);
        }

        // ---- matmul-1: a = proj + h @ W_h^T for z, r, g ----
        #pragma unroll
        for (int i = 0; i < 6; ++i) {
            int t  = wid * 6 + i;            // 0..191
            int g3 = t >> 6;                 // 0:z 1:r 2:g
            int tt = t & 63;
            int m0 = (tt & 3) << 4;
            int n0 = (tt >> 2) << 4;
            const float* W  = (g3 == 0) ? Wz : (g3 == 1) ? Wr : Wh;
            const int    pgi = (g3 == 0) ? 0 : (g3 == 1) ? 1 : 3;
            const float* pg = projs + pgi * (BATCH * HDIM);
            v8f acc;
            #pragma unroll
            for (int v = 0; v < 8; ++v)
                acc[v] = gldf(pg + (m0 + v + 8 * hi) * HDIM + n0 + l16);
            const float* arow = sh_h + (m0 + l16) * LSTR + 2 * hi;
            const float* brow = W + (n0 + l16) * WROW + IDIM + 2 * hi; // h-part
            #pragma unroll 8
            for (int k0 = 0; k0 < HDIM; k0 += 4) {
                v2f a = *(const v2f*)(arow + k0);
                v2f b = gldv2(brow + k0);
                acc = __builtin_amdgcn_wmma_f32_16x16x4_f32(false, a, false, b,
                                                            (short)0, acc, false, false);
            }
            #pragma unroll
            for (int v = 0; v < 8; ++v) {
                int idx = (m0 + v + 8 * hi) * LSTR + n0 + l16;
                float sg = sigmoid_f(acc[v]);
                if      (g3 == 0) sh_z [idx] = sg;
                else if (g3 == 1) sh_rh[idx] = sg * sh_h[idx];   // r * h
                else              sh_g [idx] = sg;
            }
        }
        __syncthreads();

        // ---- matmul-2: c = tanh(cx + (r*h) @ Wc_h^T), then combine ----
        #pragma unroll
        for (int i = 0; i < 2; ++i) {
            int tt = wid * 2 + i;            // 0..63
            int m0 = (tt & 3) << 4;
            int n0 = (tt >> 2) << 4;
            const float* pg = projs + 2 * (BATCH * HDIM);        // c slot
            v8f acc;
            #pragma unroll
            for (int v = 0; v < 8; ++v)
                acc[v] = gldf(pg + (m0 + v + 8 * hi) * HDIM + n0 + l16);
            const float* arow = sh_rh + (m0 + l16) * LSTR + 2 * hi;
            const float* brow = Wc + (n0 + l16) * WROW + IDIM + 2 * hi;
            #pragma unroll 8
            for (int k0 = 0; k0 < HDIM; k0 += 4) {
                v2f a = *(const v2f*)(arow + k0);
                v2f b = gldv2(brow + k0);
                acc = __builtin_amdgcn_wmma_f32_16x16x4_f32(false, a, false, b,
                                                            (short)0, acc, false, false);
            }
            #pragma unroll
            for (int v = 0; v < 8; ++v) {
                int idx = (m0 + v + 8 * hi) * LSTR + n0 + l16;
                float ct = tanh_f(acc[v]);
                float zv = sh_z[idx];
                float gv = sh_g[idx];
                float hv = sh_h[idx];
                float hn = (1.0f - zv) * hv + zv * ct;   // h_new (carry)
                float hf = gv * hn + (1.0f - gv) * hv;   // highway output
                sh_h[idx] = hn;                          // sole reader/writer of idx
                sh_z[idx] = hf;                          // reuse z buffer for h_final
            }
        }
        __syncthreads();

        // ---- LayerNorm over H=256 + store out[s] ----
        #pragma unroll
        for (int rr = 0; rr < 2; ++rr) {
            int row = wid * 2 + rr;
            const float* src = sh_z + row * LSTR + lane * 8;
            float v[8];
            float sum = 0.0f, ssq = 0.0f;
            #pragma unroll
            for (int j = 0; j < 8; ++j) {
                v[j] = src[j];
                sum += v[j];
                ssq += v[j] * v[j];
            }
            #pragma unroll
            for (int m = 16; m >= 1; m >>= 1) {
                sum += __shfl_xor(sum, m, 32);
                ssq += __shfl_xor(ssq, m, 32);
            }
            float mean = sum * (1.0f / 256.0f);
            float var  = ssq * (1.0f / 256.0f) - mean * mean;
            float rstd = rsqrtf(var + 1e-5f);
            float o[8];
            #pragma unroll
            for (int j = 0; j < 8; ++j)
                o[j] = (v[j] - mean) * rstd * gam[j] + bet[j];
            float* dst = out + ((size_t)s * BATCH + row) * HDIM + lane * 8;
            *(float4*)(dst)     = make_float4(o[0], o[1], o[2], o[3]);
            *(float4*)(dst + 4) = make_float4(o[4], o[5], o[6], o[7]);
        }
        __syncthreads();
    }

    // final carry h_new -> tail of d_out (shape (1,B,H))
    float* hl = out + (size_t)S_LEN * BATCH * HDIM;
    {
        int row = tid >> 4;
        int c0  = (tid & 15) * 16;
        #pragma unroll
        for (int j = 0; j < 16; ++j)
            hl[row * HDIM + c0 + j] = sh_h[row * LSTR + c0 + j];
    }
}

// ---------------------------------------------------------------------------
extern "C" void kernel_launch(void* const* d_in, const int* in_sizes, int n_in,
                              void* d_out, int out_size, void* d_ws, size_t ws_size,
                              hipStream_t stream) {
    (void)in_sizes; (void)n_in; (void)out_size; (void)ws_size;
    const float* x     = (const float*)d_in[0];
    const float* Wz    = (const float*)d_in[1];
    const float* bz    = (const float*)d_in[2];
    const float* Wr    = (const float*)d_in[3];
    const float* br    = (const float*)d_in[4];
    const float* Wc    = (const float*)d_in[5];
    const float* bc    = (const float*)d_in[6];
    const float* Wh    = (const float*)d_in[7];
    const float* bh    = (const float*)d_in[8];
    const float* gamma = (const float*)d_in[9];
    const float* beta  = (const float*)d_in[10];
    float* out  = (float*)d_out;
    float* proj = (float*)d_ws;   // S*4*B*H f32 = 512 MiB of scratch

    hgru_proj_kernel<<<S_LEN, 1024, 0, stream>>>(x, Wz, bz, Wr, br, Wc, bc, Wh, bh, proj);
    hgru_scan_kernel<<<1, 1024, 0, stream>>>(proj, Wz, Wr, Wc, Wh, gamma, beta, out);
}